// PointLoss_8796093022450
// MI455X (gfx1250) — compile-verified
//
#include <hip/hip_runtime.h>
#include <math.h>

// CDNA5 / gfx1250 — fp32 WMMA (V_WMMA_F32_16X16X4_F32) path.
typedef __attribute__((ext_vector_type(2))) float v2f;
typedef __attribute__((ext_vector_type(8))) float v8f;

#define KDIM   51           // blendshape count
#define NSTEPS 13           // ceil(51/4) WMMA K-steps (K padded to 52)
#define NCOLS  24576        // 8192 * 3, flattened (n,d) columns
#define BROWS  256
#define EPSV   1e-6f
#define MULTV  0.0025f

// ---------------------------------------------------------------------------
// Main GEMM + squared-accumulation kernel.
// grid = (16 row tiles, 96 column groups), block = 32 threads (one wave32).
// Each wave: A-tile = (y_hat - y)[16 x 52] in registers, then 16 column tiles
// of 13 x v_wmma_f32_16x16x4_f32 each, accumulating (v+EPS)^2 per (b, d=c%3).
// ---------------------------------------------------------------------------
__global__ __launch_bounds__(32) void pl_gemm_wmma(
    const float* __restrict__ y_hat,
    const float* __restrict__ y,
    const float* __restrict__ bs,     // [51][24576] row-major (== bs_tensor)
    float* __restrict__ S)            // [256][3] partial sums of squares
{
  const int lane = threadIdx.x & 31;
  const int half = lane >> 4;          // 0: lanes 0-15, 1: lanes 16-31
  const int l16  = lane & 15;
  const int rowTile  = blockIdx.x;     // 0..15  -> rows b = rowTile*16 ..
  const int colGroup = blockIdx.y;     // 0..95  -> cols c = colGroup*256 ..

  // ---- A tile: W[M][k] = y_hat - y, rows M = l16, K padded 51 -> 52 ----
  // f32 16x4 A layout: lane = M; VGPR0 holds K = 4*kk + 2*half,
  //                              VGPR1 holds K = 4*kk + 2*half + 1.
  const int m = rowTile * 16 + l16;
  const float* __restrict__ yh = y_hat + m * KDIM;
  const float* __restrict__ yy = y     + m * KDIM;
  v2f a[NSTEPS];
#pragma unroll
  for (int kk = 0; kk < NSTEPS; ++kk) {
    const int k0  = 4 * kk + 2 * half;
    const int k1  = k0 + 1;
    const int k0c = (k0 < KDIM) ? k0 : (KDIM - 1);   // clamp addr, mask value
    const int k1c = (k1 < KDIM) ? k1 : (KDIM - 1);
    const float w0 = yh[k0c] - yy[k0c];
    const float w1 = yh[k1c] - yy[k1c];
    a[kk].x = (k0 < KDIM) ? w0 : 0.0f;
    a[kk].y = (k1 < KDIM) ? w1 : 0.0f;
  }

  // per-lane accumulators, bucketed by d = column % 3
  float acc0[8], acc1[8], acc2[8];
#pragma unroll
  for (int j = 0; j < 8; ++j) { acc0[j] = 0.f; acc1[j] = 0.f; acc2[j] = 0.f; }

  for (int t = 0; t < 16; ++t) {
    const int c = colGroup * 256 + t * 16 + l16;     // this lane's column
    const float* __restrict__ bp = bs + c;
    if (t < 15) __builtin_prefetch(bp + 16, 0, 1);   // global_prefetch_b8

    v8f cc = {};
#pragma unroll
    for (int kk = 0; kk < NSTEPS; ++kk) {
      // f32 4x16 B layout: lane = N; VGPR0 holds K = 2*half, VGPR1 K = 2*half+1
      const int k0  = 4 * kk + 2 * half;
      const int k1  = k0 + 1;
      const int k0c = (k0 < KDIM) ? k0 : (KDIM - 1);
      const int k1c = (k1 < KDIM) ? k1 : (KDIM - 1);
      float b0 = bp[k0c * NCOLS];
      float b1 = bp[k1c * NCOLS];
      v2f bv;
      bv.x = (k0 < KDIM) ? b0 : 0.0f;
      bv.y = (k1 < KDIM) ? b1 : 0.0f;
      // D = A x B + C  (16x16x4, all fp32)
      cc = __builtin_amdgcn_wmma_f32_16x16x4_f32(
          /*neg_a=*/false, a[kk], /*neg_b=*/false, bv,
          /*c_mod=*/(short)0, cc, /*reuse_a=*/false, /*reuse_b=*/false);
    }

    const int d  = c % 3;
    const bool d0 = (d == 0), d1 = (d == 1);
#pragma unroll
    for (int j = 0; j < 8; ++j) {
      const float v  = cc[j] + EPSV;     // diff + eps, elementwise
      const float sq = v * v;
      acc0[j] += d0 ? sq : 0.0f;
      acc1[j] += d1 ? sq : 0.0f;
      acc2[j] += (!d0 && !d1) ? sq : 0.0f;
    }
  }

  // Butterfly-reduce across the 16 lanes of each half-wave (column partials);
  // lane^off stays inside the half for off = 8,4,2,1.
#pragma unroll
  for (int j = 0; j < 8; ++j) {
#pragma unroll
    for (int off = 8; off >= 1; off >>= 1) {
      acc0[j] += __shfl_xor(acc0[j], off, 32);
      acc1[j] += __shfl_xor(acc1[j], off, 32);
      acc2[j] += __shfl_xor(acc2[j], off, 32);
    }
  }

  // C/D layout: VGPR j, half h -> row M = j + 8*h.
  if (l16 == 0) {
    const int b = rowTile * 16 + half * 8;
#pragma unroll
    for (int j = 0; j < 8; ++j) {
      atomicAdd(&S[(b + j) * 3 + 0], acc0[j]);
      atomicAdd(&S[(b + j) * 3 + 1], acc1[j]);
      atomicAdd(&S[(b + j) * 3 + 2], acc2[j]);
    }
  }
}

// Zero the 768-float workspace (harness poisons it with 0xAA).
__global__ __launch_bounds__(256) void pl_zero(float* __restrict__ S) {
  const int i = blockIdx.x * 256 + threadIdx.x;
  if (i < BROWS * 3) S[i] = 0.0f;
}

// Final reduction: out = MULT * sum_{b,d} sqrt(S[b][d]).
__global__ __launch_bounds__(256) void pl_finish(const float* __restrict__ S,
                                                 float* __restrict__ out) {
  __shared__ float red[256];
  const int t = threadIdx.x;                    // t == b, 0..255
  const float s = sqrtf(S[t * 3 + 0]) + sqrtf(S[t * 3 + 1]) + sqrtf(S[t * 3 + 2]);
  red[t] = s;
  __syncthreads();
#pragma unroll
  for (int stride = 128; stride > 0; stride >>= 1) {
    if (t < stride) red[t] += red[t + stride];
    __syncthreads();
  }
  if (t == 0) out[0] = red[0] * MULTV;
}

extern "C" void kernel_launch(void* const* d_in, const int* in_sizes, int n_in,
                              void* d_out, int out_size, void* d_ws, size_t ws_size,
                              hipStream_t stream) {
  (void)in_sizes; (void)n_in; (void)out_size; (void)ws_size;
  const float* y_hat = (const float*)d_in[0];   // [256,51]
  const float* y     = (const float*)d_in[1];   // [256,51]
  /* d_in[2] = face [8192,3] — cancels exactly in the difference */
  const float* bs    = (const float*)d_in[3];   // [51,8192,3] == [51,24576]
  float* S   = (float*)d_ws;                    // 768 floats of scratch
  float* out = (float*)d_out;                   // single f32 scalar

  pl_zero<<<dim3((BROWS * 3 + 255) / 256), 256, 0, stream>>>(S);
  pl_gemm_wmma<<<dim3(16, 96), 32, 0, stream>>>(y_hat, y, bs, S);
  pl_finish<<<1, 256, 0, stream>>>(S, out);
}